// RoIPooling2D_74148315398539
// MI455X (gfx1250) — compile-verified
//
#include <hip/hip_runtime.h>
#include <stdint.h>

// Problem constants (from reference setup_inputs)
#define NN 4
#define CC 256
#define HH 50
#define WW 50
#define PH 7
#define PW 7
#define CB 8                     // channels per TDM batch
#define MAXTILE (49 * 49)        // max per-channel tile elems (tile ends are clipped to <=49)
#define BUFE (CB * MAXTILE)      // 19208 floats per buffer

typedef unsigned int uv4 __attribute__((ext_vector_type(4)));
typedef int          iv8 __attribute__((ext_vector_type(8)));
typedef int          iv4 __attribute__((ext_vector_type(4)));

__device__ __forceinline__ int imin(int a, int b) { return a < b ? a : b; }
__device__ __forceinline__ int imax(int a, int b) { return a > b ? a : b; }

// Issue one TDM DMA: 3D tile (x:wW, y:wH, z:nz channels) of f32 from global
// (strides W, H*W elements) into LDS at byte offset lds_off, packed
// contiguously as [z][y][x].
__device__ __forceinline__ void tdm_load_tile(const float* gptr, unsigned lds_off,
                                              int wW, int wH, int nz) {
  unsigned long long ga = (unsigned long long)(uintptr_t)gptr;
  uv4 g0;
  g0[0] = 1u;                                            // count=1 valid descriptor
  g0[1] = lds_off;                                       // lds_addr (bytes)
  g0[2] = (unsigned)(ga & 0xFFFFFFFFull);                // global_addr[31:0]
  g0[3] = (unsigned)((ga >> 32) & 0x01FFFFFFull)         // global_addr[56:32]
        | 0x80000000u;                                   // type=2 ("image")
  const unsigned td0 = 1u << 20, td1 = 1u << 20, td2 = 1u << 20; // oversized dims: OOB never fires
  const unsigned s0 = WW;            // tensor_dim0_stride (elements)
  const unsigned s1 = HH * WW;       // tensor_dim1_stride (elements)
  iv8 g1;
  g1[0] = (int)(2u << 16);                               // data_size=2 -> 4 bytes; mask/flags=0
  g1[1] = (int)((td0 & 0xFFFFu) << 16);                  // tensor_dim0[15:0]
  g1[2] = (int)((td0 >> 16) | ((td1 & 0xFFFFu) << 16));  // tensor_dim0 hi | tensor_dim1 lo
  g1[3] = (int)((td1 >> 16) | ((unsigned)wW << 16));     // tensor_dim1 hi | tile_dim0
  g1[4] = (int)((unsigned)wH | ((unsigned)nz << 16));    // tile_dim1 | tile_dim2
  g1[5] = (int)s0;                                       // tensor_dim0_stride[31:0]
  g1[6] = (int)((s1 & 0xFFFFu) << 16);                   // stride0 hi16=0 | stride1[15:0]
  g1[7] = (int)(s1 >> 16);                               // tensor_dim1_stride[47:16]
  iv4 g2;
  g2[0] = (int)td2;                                      // tensor_dim2
  g2[1] = 0; g2[2] = 0; g2[3] = 0;                       // tile_dim3=0 (unused)
  iv4 g3; g3[0] = 0; g3[1] = 0; g3[2] = 0; g3[3] = 0;    // dims 4 unused
  iv8 g4;                                                // extra group (6-arg builtin form): zeros
  g4[0] = 0; g4[1] = 0; g4[2] = 0; g4[3] = 0;
  g4[4] = 0; g4[5] = 0; g4[6] = 0; g4[7] = 0;
  __builtin_amdgcn_tensor_load_to_lds(g0, g1, g2, g3, g4, 0);
}

__global__ __launch_bounds__(256) void roipool_tdm_kernel(
    const float* __restrict__ feat, const float* __restrict__ rois,
    float* __restrict__ out) {
  __shared__ int sXS[PW], sXE[PW], sYS[PH], sYE[PH];
  __shared__ int sMeta[5];           // b, tx0, ty0, wW, wH
  __shared__ float buf[2][BUFE];

  const int r = blockIdx.x;
  const int tid = threadIdx.x;

  if (tid < PH) {
    const float* roi = rois + (size_t)r * 5;
    int b  = (int)roi[0];
    int x0 = (int)rintf(roi[1] * (1.0f / 32.0f));   // rintf == jnp.round (ties-to-even)
    int y0 = (int)rintf(roi[2] * (1.0f / 32.0f));
    int x1 = (int)rintf(roi[3] * (1.0f / 32.0f));
    int y1 = (int)rintf(roi[4] * (1.0f / 32.0f));
    float bw = (float)imax(x1 - x0 + 1, 1);
    float bh = (float)imax(y1 - y0 + 1, 1);
    float binw = bw / (float)PW;
    float binh = bh / (float)PH;
    int i = tid;
    sYS[i] = imin(imax(y0 + (int)floorf((float)i * binh), 0), HH - 1);
    sYE[i] = imin(imax(y0 + (int)ceilf((float)(i + 1) * binh), 0), HH - 1);
    sXS[i] = imin(imax(x0 + (int)floorf((float)i * binw), 0), WW - 1);
    sXE[i] = imin(imax(x0 + (int)ceilf((float)(i + 1) * binw), 0), WW - 1);
    if (i == 0) {
      int tx0 = imin(imax(x0, 0), WW - 1);
      int ty0 = imin(imax(y0, 0), HH - 1);
      int tx1 = imin(imax(x0 + (int)ceilf((float)PW * binw), 0), WW - 1);
      int ty1 = imin(imax(y0 + (int)ceilf((float)PH * binh), 0), HH - 1);
      sMeta[0] = b;
      sMeta[1] = tx0;
      sMeta[2] = ty0;
      sMeta[3] = tx1 - tx0;   // wW
      sMeta[4] = ty1 - ty0;   // wH
    }
  }
  __syncthreads();

  // Wave-uniform roi metadata (force SGPRs for the TDM descriptor build)
  const int b   = __builtin_amdgcn_readfirstlane(sMeta[0]);
  const int tx0 = __builtin_amdgcn_readfirstlane(sMeta[1]);
  const int ty0 = __builtin_amdgcn_readfirstlane(sMeta[2]);
  const int wW  = __builtin_amdgcn_readfirstlane(sMeta[3]);
  const int wH  = __builtin_amdgcn_readfirstlane(sMeta[4]);

  float* outr = out + (size_t)r * CC * PH * PW;

  if (wW <= 0 || wH <= 0) {          // degenerate roi: every bin invalid -> zeros
    for (int u = tid; u < CC * PH * PW; u += 256) outr[u] = 0.0f;
    return;
  }

  const float* gbase = feat + ((size_t)(b * CC) * HH + ty0) * WW + tx0;
  const unsigned lds0 = (unsigned)(uintptr_t)(void*)&buf[0][0];
  const unsigned lds1 = (unsigned)(uintptr_t)(void*)&buf[1][0];

  const int NB = CC / CB;            // 32 channel batches
  const bool issuer = (tid < 32);    // wave 0 only issues TDM ops

  if (issuer) tdm_load_tile(gbase, lds0, wW, wH, CB);   // prologue: batch 0 -> buf0

  for (int k = 0; k < NB; ++k) {
    const float* lbuf = buf[k & 1];
    if (issuer) {
      if (k + 1 < NB) {
        // Prefetch next batch into the other buffer (its last reader finished
        // at the end-of-iteration barrier of k-1), then wait for batch k.
        tdm_load_tile(gbase + (size_t)(k + 1) * CB * HH * WW,
                      (k & 1) ? lds0 : lds1, wW, wH, CB);
        __builtin_amdgcn_s_wait_tensorcnt((short)1);
      } else {
        __builtin_amdgcn_s_wait_tensorcnt((short)0);
      }
    }
    __syncthreads();                 // batch k data visible to all 8 waves

    const int c0 = k * CB;
    for (int u = tid; u < CB * PH * PW; u += 256) {
      const int cl = u / (PH * PW);
      const int ij = u % (PH * PW);
      const int i = ij / PW, j = ij % PW;
      const int ys = sYS[i], ye = sYE[i], xs = sXS[j], xe = sXE[j];
      const bool valid = (ye > ys) && (xe > xs);
      float m = -__builtin_huge_valf();
      if (valid) {
        const float* cbase = lbuf + (size_t)cl * wH * wW;
        for (int y = ys; y < ye; ++y) {
          const float* row = cbase + (size_t)(y - ty0) * wW - tx0;
          for (int x = xs; x < xe; ++x) m = fmaxf(m, row[x]);
        }
      }
      outr[(size_t)(c0 + cl) * (PH * PW) + ij] = valid ? m : 0.0f;
    }
    __syncthreads();                 // all reads of buf[k&1] done before it is reused
  }
}

extern "C" void kernel_launch(void* const* d_in, const int* in_sizes, int n_in,
                              void* d_out, int out_size, void* d_ws, size_t ws_size,
                              hipStream_t stream) {
  const float* feat = (const float*)d_in[0];
  const float* rois = (const float*)d_in[1];
  float* out = (float*)d_out;
  const int R = in_sizes[1] / 5;     // 256
  roipool_tdm_kernel<<<dim3(R), dim3(256), 0, stream>>>(feat, rois, out);
}